// DPB_24026047054525
// MI455X (gfx1250) — compile-verified
//
#include <hip/hip_runtime.h>

// Native ext-vector types (HIP's float4 is a struct; builtins want real vectors)
typedef float v2f __attribute__((ext_vector_type(2)));
typedef float v4f __attribute__((ext_vector_type(4)));
typedef float v8f __attribute__((ext_vector_type(8)));

#define LSTR 65   // LDS row stride (floats) — padded to dodge 16-way bank conflicts

__device__ __forceinline__ float silu_f(float x) {
    return x / (1.0f + expf(-x));
}

// One MLP layer: dst(16x64) = silu( src(16x64) @ W(64x64) + bias ), via
// V_WMMA_F32_16X16X4_F32. Block = 128 threads = 4 waves; wave w owns the
// 16-column tile [16w, 16w+16). K-loop: 16 steps of K=4.
__device__ __forceinline__ void mlp_layer_wmma(const float* __restrict__ srcLds,
                                               const float* __restrict__ W,
                                               const float* __restrict__ bias,
                                               float* __restrict__ dstLds)
{
    const int lane  = threadIdx.x & 31;
    const int wave  = threadIdx.x >> 5;     // 0..3 -> N tile
    const int mrow  = lane & 15;            // A: M index / B,D: N index (lane%16)
    const int hi    = lane >> 4;            // lane-half select
    const int khalf = hi * 2;               // K offset within 4-wide K step
    const int ncol  = wave * 16 + mrow;     // global output column

    v8f acc = {};
    #pragma unroll
    for (int k0 = 0; k0 < 64; k0 += 4) {
        v2f a, b;
        // A 16x4 fp32 layout: VGPR0 = K(0|2), VGPR1 = K(1|3) per lane-half
        a.x = srcLds[mrow * LSTR + k0 + khalf];
        a.y = srcLds[mrow * LSTR + k0 + khalf + 1];
        // B 4x16 fp32 layout (symmetric striping): lane holds column ncol
        b.x = W[(k0 + khalf) * 64 + ncol];
        b.y = W[(k0 + khalf + 1) * 64 + ncol];
        acc = __builtin_amdgcn_wmma_f32_16x16x4_f32(
            /*neg_a=*/false, a, /*neg_b=*/false, b,
            /*c_mod=*/(short)0, acc, /*reuse_a=*/false, /*reuse_b=*/false);
    }
    // D 16x16 fp32: VGPR r, lane l -> M = r + 8*(l>=16), N = l&15
    #pragma unroll
    for (int r = 0; r < 8; ++r) {
        int M = r + hi * 8;
        dstLds[M * LSTR + ncol] = silu_f(acc[r] + bias[ncol]);
    }
}

// Build tt[8][2048]: tt[h][d] = MLP(d - 1023)[h] for d = 0..2047.
// Grid: 128 blocks x 128 threads; block handles 16 consecutive deltas.
__global__ void __launch_bounds__(128)
build_bias_table(const float* __restrict__ W0, const float* __restrict__ b0,
                 const float* __restrict__ W1, const float* __restrict__ b1,
                 const float* __restrict__ W2, const float* __restrict__ b2,
                 const float* __restrict__ Wf, const float* __restrict__ bf,
                 float* __restrict__ tt)
{
    __shared__ float bufA[16 * LSTR];
    __shared__ float bufB[16 * LSTR];
    const int m0 = blockIdx.x * 16;

    // Layer 0: h = silu(delta * W0 + b0)   (W0 is 1x64 -> outer product w/ scalar)
    for (int e = threadIdx.x; e < 16 * 64; e += 128) {
        int row = e >> 6, n = e & 63;
        float delta = (float)(m0 + row - 1023);
        bufA[row * LSTR + n] = silu_f(delta * W0[n] + b0[n]);
    }
    __syncthreads();

    mlp_layer_wmma(bufA, W1, b1, bufB);   // layer 1 (WMMA)
    __syncthreads();
    mlp_layer_wmma(bufB, W2, b2, bufA);   // layer 2 (WMMA)
    __syncthreads();

    // Final 64->8 projection: 16x8 outputs, one per thread
    for (int e = threadIdx.x; e < 16 * 8; e += 128) {
        int row = e >> 3, hc = e & 7;
        float acc = bf[hc];
        #pragma unroll
        for (int k = 0; k < 64; ++k)
            acc += bufA[row * LSTR + k] * Wf[k * 8 + hc];
        tt[hc * 2048 + (m0 + row)] = acc;   // transposed layout: [h][delta]
    }
}

// out[b,h,i,j] = mask[b,h,i,j] - tt[h][i-j+1023]
// Grid: 2048 blocks x 256 threads. Block = 8 i-rows of one (b,h) plane.
// Pure HBM stream: 64 MB read + 64 MB write, NT-hinted both ways.
__global__ void __launch_bounds__(256)
apply_bias(const float* __restrict__ mask, const float* __restrict__ tt,
           float* __restrict__ out)
{
    __shared__ float trow[2048];
    const int blk = blockIdx.x;        // 0..2047
    const int ib  = blk & 127;         // i-block (8 rows)
    const int bh  = blk >> 7;          // 0..15 = b*8 + h
    const int h   = bh & 7;

    for (int e = threadIdx.x; e < 2048; e += 256)
        trow[e] = tt[h * 2048 + e];
    __syncthreads();

    const size_t planeOff = ((size_t)bh * 1024 + (size_t)ib * 8) * 1024;
    const v4f* __restrict__ mrow = (const v4f*)(mask + planeOff);
    v4f*       __restrict__ orow = (v4f*)(out + planeOff);

    #pragma unroll
    for (int it = 0; it < 8; ++it) {
        int idx4 = it * 256 + (int)threadIdx.x;   // 0..2047 float4s
        int row  = idx4 >> 8;                     // 0..7
        int j    = (idx4 & 255) * 4;
        int i    = ib * 8 + row;
        int d0   = i - j + 1023;                  // in [3, 2046] for j..j+3 valid
        v4f m = __builtin_nontemporal_load(mrow + idx4);
        v4f o;
        o.x = m.x - trow[d0];
        o.y = m.y - trow[d0 - 1];
        o.z = m.z - trow[d0 - 2];
        o.w = m.w - trow[d0 - 3];
        __builtin_nontemporal_store(o, orow + idx4);
    }
}

extern "C" void kernel_launch(void* const* d_in, const int* in_sizes, int n_in,
                              void* d_out, int out_size, void* d_ws, size_t ws_size,
                              hipStream_t stream) {
    // Input order: mask, q, k, v, W0, b0, W1, b1, W2, b2, Wf, bf (q/k/v unused)
    const float* mask = (const float*)d_in[0];
    const float* W0   = (const float*)d_in[4];
    const float* b0v  = (const float*)d_in[5];
    const float* W1   = (const float*)d_in[6];
    const float* b1v  = (const float*)d_in[7];
    const float* W2   = (const float*)d_in[8];
    const float* b2v  = (const float*)d_in[9];
    const float* Wf   = (const float*)d_in[10];
    const float* bfv  = (const float*)d_in[11];
    float* out = (float*)d_out;
    float* tt  = (float*)d_ws;   // 8 * 2048 floats = 64 KB scratch

    build_bias_table<<<128, 128, 0, stream>>>(W0, b0v, W1, b1v, W2, b2v, Wf, bfv, tt);
    apply_bias<<<2048, 256, 0, stream>>>(mask, tt, out);
}